// SpatialOut_44057774522753
// MI455X (gfx1250) — compile-verified
//
#include <hip/hip_runtime.h>

typedef __attribute__((ext_vector_type(2))) float v2f;
typedef __attribute__((ext_vector_type(8))) float v8f;

#define NODE_DIM 128
#define HID 64

// ---------------- Pass 1: mass-weighted segment sums -> acc[B][4] ----------
__global__ void centroid_accum(const float* __restrict__ pos,
                               const float* __restrict__ masses,
                               const int* __restrict__ batch,
                               const int* __restrict__ anum,
                               float* __restrict__ acc, int nAtoms) {
    for (int i = blockIdx.x * blockDim.x + threadIdx.x; i < nAtoms;
         i += gridDim.x * blockDim.x) {
        float m = masses[anum[i]];
        int b = batch[i];
        float x = pos[i * 3 + 0];
        float y = pos[i * 3 + 1];
        float z = pos[i * 3 + 2];
        atomicAdd(&acc[b * 4 + 0], m * x);
        atomicAdd(&acc[b * 4 + 1], m * y);
        atomicAdd(&acc[b * 4 + 2], m * z);
        atomicAdd(&acc[b * 4 + 3], m);
    }
}

// ---------------- Pass 2: centroids[B][4] = acc.xyz / acc.w ----------------
__global__ void centroid_finalize(const float* __restrict__ acc,
                                  float* __restrict__ cent, int bseg) {
    int b = blockIdx.x * blockDim.x + threadIdx.x;
    if (b < bseg) {
        float inv = 1.0f / acc[b * 4 + 3];
        cent[b * 4 + 0] = acc[b * 4 + 0] * inv;
        cent[b * 4 + 1] = acc[b * 4 + 1] * inv;
        cent[b * 4 + 2] = acc[b * 4 + 2] * inv;
        cent[b * 4 + 3] = 0.0f;
    }
}

// ---------------- Pass 3: WMMA MLP + spatial extent scatter ----------------
// One wave32 per 16-atom tile. h[16,64] = A[16,128] x W1^T[128,64] using
// v_wmma_f32_16x16x4_f32: 32 K-steps x 4 N-tiles.
// LDS holds W1^T as float2 K-pairs: ldsB[kpair*65 + h] = (w1T[2p][h], w1T[2p+1][h]).
__global__ __launch_bounds__(256) void mlp_extent(
    const float* __restrict__ pos, const float* __restrict__ ns,
    const float* __restrict__ w1, const float* __restrict__ b1,
    const float* __restrict__ w2, const float* __restrict__ b2,
    const int* __restrict__ batch, const float* __restrict__ cent,
    float* __restrict__ out, int nAtoms, int numTiles) {
    __shared__ v2f ldsB[64 * 65];  // 64 kpairs x 64 hid, stride 65 (pad)

    const int t = threadIdx.x;
    // Stage w1 (row-major [64][128]) into LDS as w1T float2 pairs.
    const v2f* w1p = (const v2f*)w1;  // [64][64] float2, pair = (c,c+1)
    #pragma unroll
    for (int e = 0; e < 16; ++e) {
        int idx = e * 256 + t;       // 0..4095
        int p = idx & 63;            // kpair (varies fastest -> coalesced)
        int h = idx >> 6;            // hid
        ldsB[p * 65 + h] = w1p[h * 64 + p];
    }
    __syncthreads();

    const int lane = t & 31;
    const int warp = t >> 5;
    const int nlo = lane & 15;   // col within 16 / atom row for A
    const int hi = lane >> 4;    // half-wave select (K split)
    const int waveId = blockIdx.x * 8 + warp;
    const int waveStride = gridDim.x * 8;
    const float bias2 = b2[0];

    // Per-lane MLP constants (tile-invariant): column n = nt*16 + nlo.
    float w2n[4], b1n[4];
    #pragma unroll
    for (int nt = 0; nt < 4; ++nt) {
        w2n[nt] = w2[nt * 16 + nlo];
        b1n[nt] = b1[nt * 16 + nlo];
    }

    for (int tile = waveId; tile < numTiles; tile += waveStride) {
        const int a0 = tile * 16;
        // Fold b1 into the accumulators: D = A*B + C starts at the bias.
        v8f c0, c1, c2, c3;
        #pragma unroll
        for (int j = 0; j < 8; ++j) {
            c0[j] = b1n[0]; c1[j] = b1n[1]; c2[j] = b1n[2]; c3[j] = b1n[3];
        }

        // A-frag base: lane holds node_scalar[a0+nlo][k0 + 2*hi .. +1]
        const float* arow = ns + (size_t)(a0 + nlo) * NODE_DIM + 2 * hi;

        // Prefetch next tile's A rows (64 cachelines across the wave).
        if (tile + waveStride < numTiles) {
            const float* pf =
                ns + (size_t)((tile + waveStride) * 16 + nlo) * NODE_DIM + hi * 64;
            __builtin_prefetch(pf, 0, 0);
            __builtin_prefetch(pf + 32, 0, 0);
        }

        #pragma unroll 4
        for (int ks = 0; ks < 32; ++ks) {  // k0 = 4*ks
            v2f a = *(const v2f*)(arow + 4 * ks);
            const v2f* brow = &ldsB[(2 * ks + hi) * 65 + nlo];
            c0 = __builtin_amdgcn_wmma_f32_16x16x4_f32(false, a, false, brow[0],
                                                       (short)0, c0, false, false);
            c1 = __builtin_amdgcn_wmma_f32_16x16x4_f32(false, a, false, brow[16],
                                                       (short)0, c1, false, false);
            c2 = __builtin_amdgcn_wmma_f32_16x16x4_f32(false, a, false, brow[32],
                                                       (short)0, c2, false, false);
            c3 = __builtin_amdgcn_wmma_f32_16x16x4_f32(false, a, false, brow[48],
                                                       (short)0, c3, false, false);
        }

        // Epilogue: s[j] = sum_n w2[n]*silu(h[row][n]) partials.
        // C layout: VGPR j, lane L -> row (j + 8*(L>=16)), col (L&15)+16*nt.
        float s[8];
        #pragma unroll
        for (int j = 0; j < 8; ++j) s[j] = 0.0f;

        auto fold = [&](const v8f& cc, int nt) {
            #pragma unroll
            for (int j = 0; j < 8; ++j) {
                float x = cc[j];  // bias already accumulated
                // Fast SiLU: x * rcp(1 + exp(-x)) -> v_exp_f32 + v_rcp_f32
                float sig = __builtin_amdgcn_rcpf(1.0f + __expf(-x));
                s[j] += w2n[nt] * (x * sig);
            }
        };
        fold(c0, 0); fold(c1, 1); fold(c2, 2); fold(c3, 3);

        // Reduce across the 16 lanes of each half-wave (masks 1..8 stay in-half).
        #pragma unroll
        for (int j = 0; j < 8; ++j) {
            float v = s[j];
            v += __shfl_xor(v, 1);
            v += __shfl_xor(v, 2);
            v += __shfl_xor(v, 4);
            v += __shfl_xor(v, 8);
            s[j] = v;
        }

        // Lanes 0..7 handle rows 0..7; lanes 16..23 handle rows 8..15.
        if (nlo < 8) {
            float val = s[0];
            #pragma unroll
            for (int j = 1; j < 8; ++j) val = (nlo == j) ? s[j] : val;
            int m = nlo + 8 * hi;
            int a = a0 + m;
            if (a < nAtoms) {
                int b = batch[a];
                float dx = pos[a * 3 + 0] - cent[b * 4 + 0];
                float dy = pos[a * 3 + 1] - cent[b * 4 + 1];
                float dz = pos[a * 3 + 2] - cent[b * 4 + 2];
                float spatial = dx * dx + dy * dy + dz * dz;
                atomicAdd(&out[b], (val + bias2) * spatial);
            }
        }
    }
}

extern "C" void kernel_launch(void* const* d_in, const int* in_sizes, int n_in,
                              void* d_out, int out_size, void* d_ws, size_t ws_size,
                              hipStream_t stream) {
    const float* pos    = (const float*)d_in[0];
    const float* ns     = (const float*)d_in[1];
    const float* masses = (const float*)d_in[2];
    const float* w1     = (const float*)d_in[3];
    const float* b1     = (const float*)d_in[4];
    const float* w2     = (const float*)d_in[5];
    const float* b2     = (const float*)d_in[6];
    const int*   batch  = (const int*)d_in[7];
    const int*   anum   = (const int*)d_in[8];
    float* out = (float*)d_out;

    const int nAtoms = in_sizes[7];  // N (batch array length)
    const int bseg   = out_size;     // B = 8192

    float* acc   = (float*)d_ws;                // [B][4]
    float* centr = acc + (size_t)bseg * 4;      // [B][4]

    hipMemsetAsync(acc, 0, (size_t)bseg * 4 * sizeof(float), stream);
    hipMemsetAsync(out, 0, (size_t)bseg * sizeof(float), stream);

    centroid_accum<<<2048, 256, 0, stream>>>(pos, masses, batch, anum, acc, nAtoms);
    centroid_finalize<<<(bseg + 255) / 256, 256, 0, stream>>>(acc, centr, bseg);

    int numTiles = (nAtoms + 15) / 16;
    int blocks = (numTiles + 7) / 8;  // 8 waves per block, 1 tile per wave pass
    mlp_extent<<<blocks, 256, 0, stream>>>(pos, ns, w1, b1, w2, b2, batch, centr,
                                           out, nAtoms, numTiles);
}